// GraphAttentionLayer_45277545234539
// MI455X (gfx1250) — compile-verified
//
#include <hip/hip_runtime.h>
#include <hip/hip_bf16.h>

#define IN_F   256
#define OUT_F  128
#define ALPHA  0.2f
#define A_PAD  (IN_F + 4)   // 260 floats per LDS row -> conflict-free wmma fragment reads

typedef float v2f __attribute__((ext_vector_type(2)));
typedef float v8f __attribute__((ext_vector_type(8)));

// ---------- ordered-float encoding so float max == uint max ----------
__device__ __forceinline__ unsigned enc_ordered(float x) {
    unsigned b = __float_as_uint(x);
    return (b & 0x80000000u) ? ~b : (b | 0x80000000u);
}
__device__ __forceinline__ float dec_ordered(unsigned u) {
    unsigned b = (u & 0x80000000u) ? (u & 0x7FFFFFFFu) : ~u;
    return __uint_as_float(b);
}

// native f32 atomic add (GLOBAL_ATOMIC_ADD_F32, no CAS loop)
__device__ __forceinline__ void fadd_atomic(float* p, float v) {
    unsafeAtomicAdd(p, v);
}

// ---------- 0: init out accumulator + per-node max/denom ----------
__global__ void gat_init(float* __restrict__ out, int nOut,
                         unsigned* __restrict__ mmax, float* __restrict__ denom, int nNode) {
    int i = blockIdx.x * blockDim.x + threadIdx.x;
    if (i < nOut)  out[i] = 0.0f;
    if (i < nNode) { mmax[i] = 0u; denom[i] = 0.0f; }   // enc 0 = smallest key
}

// ---------- 1: Wh = h @ W via v_wmma_f32_16x16x4_f32 ----------
__launch_bounds__(256)
__global__ void gat_gemm(const float* __restrict__ h, const float* __restrict__ W,
                         float* __restrict__ Wh, int nnode) {
    __shared__ __align__(16) float As[16][A_PAD];      // 16 rows of h tile
    __shared__ __align__(16) float Bt[OUT_F][A_PAD];   // W transposed: Bt[n][k]

    const int tid  = threadIdx.x;
    const int wave = tid >> 5;
    const int lane = tid & 31;
    const int m0   = blockIdx.x * 16;
    const bool full = (m0 + 16 <= nnode);

    // Load W transposed into LDS (coalesced global reads).
    for (int i = tid; i < IN_F * OUT_F; i += 256) {
        int k = i >> 7;        // row of W (K index)
        int n = i & 127;       // col of W (N index)
        Bt[n][k] = W[i];
    }
    // Load 16 x 256 A tile (float4 per thread x4).
    if (full) {
        #pragma unroll
        for (int i = tid * 4; i < 16 * IN_F; i += 256 * 4) {
            int r = i >> 8;
            int c = i & 255;
            *(float4*)&As[r][c] = *(const float4*)&h[(size_t)(m0 + r) * IN_F + c];
        }
    } else {
        for (int i = tid * 4; i < 16 * IN_F; i += 256 * 4) {
            int r = i >> 8;
            int c = i & 255;
            int gm = m0 + r;
            float4 v = make_float4(0.f, 0.f, 0.f, 0.f);
            if (gm < nnode) v = *(const float4*)&h[(size_t)gm * IN_F + c];
            *(float4*)&As[r][c] = v;
        }
    }
    __syncthreads();

    const int lm = lane & 15;     // M (for A) / N (for B) within tile
    const int kh = lane >> 4;     // K-half selector
    const int n0 = wave * 16;

    v8f acc = {};
    #pragma unroll 8
    for (int k0 = 0; k0 < IN_F; k0 += 4) {
        v2f a, b;
        // A 16x4: lanes 0-15 -> K=0,1 ; lanes 16-31 -> K=2,3
        a.x = As[lm][k0 + 2 * kh];
        a.y = As[lm][k0 + 2 * kh + 1];
        // B 4x16: same K striping, N across lanes
        b.x = Bt[n0 + lm][k0 + 2 * kh];
        b.y = Bt[n0 + lm][k0 + 2 * kh + 1];
        acc = __builtin_amdgcn_wmma_f32_16x16x4_f32(
            /*neg_a=*/false, a, /*neg_b=*/false, b,
            /*c_mod=*/(short)0, acc, /*reuse_a=*/false, /*reuse_b=*/false);
    }

    // D layout: lanes 0-15 hold M=0..7 (vgpr r), lanes 16-31 hold M=8..15
    const int ncol  = n0 + lm;
    const int mbase = m0 + kh * 8;
    if (full) {
        #pragma unroll
        for (int r = 0; r < 8; ++r)
            Wh[(size_t)(mbase + r) * OUT_F + ncol] = acc[r];
    } else {
        #pragma unroll
        for (int r = 0; r < 8; ++r) {
            int gm = mbase + r;
            if (gm < nnode) Wh[(size_t)gm * OUT_F + ncol] = acc[r];
        }
    }
}

// ---------- 2: Wh1 = Wh @ a[:128], Wh2 = Wh @ a[128:] ----------
__global__ void gat_proj(const float* __restrict__ Wh, const float* __restrict__ a,
                         float* __restrict__ Wh1, float* __restrict__ Wh2, int n) {
    int i = blockIdx.x * blockDim.x + threadIdx.x;
    if (i >= n) return;
    const float4* row = (const float4*)(Wh + (size_t)i * OUT_F);
    const float4* a1  = (const float4*)a;
    const float4* a2  = (const float4*)(a + OUT_F);
    float s1 = 0.f, s2 = 0.f;
    #pragma unroll
    for (int f = 0; f < OUT_F / 4; ++f) {
        float4 v  = row[f];
        float4 c1 = a1[f];
        float4 c2 = a2[f];
        s1 += v.x * c1.x + v.y * c1.y + v.z * c1.z + v.w * c1.w;
        s2 += v.x * c2.x + v.y * c2.y + v.z * c2.z + v.w * c2.w;
    }
    Wh1[i] = s1;
    Wh2[i] = s2;
}

// ---------- 3: per-edge logits + segment max (src) ----------
__global__ void gat_logits(const long long* __restrict__ src, const long long* __restrict__ dst,
                           const float* __restrict__ Wh1, const float* __restrict__ Wh2,
                           float* __restrict__ ebuf, unsigned* __restrict__ mmax, int nedge) {
    int i = blockIdx.x * blockDim.x + threadIdx.x;
    if (i >= nedge) return;
    int s = (int)src[i];
    int d = (int)dst[i];
    float v = Wh1[s] + Wh2[d];
    v = v > 0.f ? v : ALPHA * v;          // LeakyReLU
    ebuf[i] = v;
    atomicMax(&mmax[s], enc_ordered(v));
}

// ---------- 4: exp(e - max) + segment sum (src); ex written in place ----------
__global__ void gat_exp(const long long* __restrict__ src,
                        float* __restrict__ ebuf, const unsigned* __restrict__ mmax,
                        float* __restrict__ denom, int nedge) {
    int i = blockIdx.x * blockDim.x + threadIdx.x;
    if (i >= nedge) return;
    int s = (int)src[i];
    float m = dec_ordered(mmax[s]);
    float x = __expf(ebuf[i] - m);
    ebuf[i] = x;
    fadd_atomic(&denom[s], x);
}

// ---------- 5: scatter-accumulate ex * Wh[dst] into out[src]; 1 wave / edge ----------
__launch_bounds__(256)
__global__ void gat_agg(const long long* __restrict__ src, const long long* __restrict__ dst,
                        const float* __restrict__ ex, const float* __restrict__ Wh,
                        float* __restrict__ out, int nedge) {
    int gt   = blockIdx.x * blockDim.x + threadIdx.x;
    // edge index is wave-uniform: force it into an SGPR so src/dst/ex become scalar loads
    int e    = __builtin_amdgcn_readfirstlane(gt >> 5);
    int lane = threadIdx.x & 31;
    if (e >= nedge) return;
    int s = (int)src[e];
    int d = (int)dst[e];
    float w = ex[e];
    const float4 v = *(const float4*)&Wh[(size_t)d * OUT_F + lane * 4];
    float* o = out + (size_t)s * OUT_F + lane * 4;
    fadd_atomic(o + 0, w * v.x);
    fadd_atomic(o + 1, w * v.y);
    fadd_atomic(o + 2, w * v.z);
    fadd_atomic(o + 3, w * v.w);
}

// ---------- 6: divide by denom + ELU ----------
__global__ void gat_final(float* __restrict__ out, const float* __restrict__ denom, int total) {
    int i = blockIdx.x * blockDim.x + threadIdx.x;
    if (i >= total) return;
    float den = denom[i >> 7];           // / OUT_F
    float v   = out[i];
    float r   = den > 0.f ? v / den : 0.f;
    out[i] = r > 0.f ? r : (__expf(r) - 1.f);   // ELU(alpha=1)
}

extern "C" void kernel_launch(void* const* d_in, const int* in_sizes, int n_in,
                              void* d_out, int out_size, void* d_ws, size_t ws_size,
                              hipStream_t stream) {
    const float*     h  = (const float*)d_in[0];
    const float*     W  = (const float*)d_in[1];
    const float*     a  = (const float*)d_in[2];
    const long long* ei = (const long long*)d_in[3];

    float* out = (float*)d_out;

    const int nnode = in_sizes[0] / IN_F;
    const int nedge = in_sizes[3] / 2;
    const long long* src = ei;
    const long long* dst = ei + nedge;

    // workspace layout
    float*    Wh    = (float*)d_ws;                       // nnode * OUT_F
    float*    Wh1   = Wh  + (size_t)nnode * OUT_F;        // nnode
    float*    Wh2   = Wh1 + nnode;                        // nnode
    float*    denom = Wh2 + nnode;                        // nnode
    unsigned* mmax  = (unsigned*)(denom + nnode);         // nnode
    float*    ebuf  = (float*)(mmax + nnode);             // nedge (e, then ex in place)

    const int nOut = nnode * OUT_F;

    gat_init  <<<(nOut + 255) / 256, 256, 0, stream>>>(out, nOut, mmax, denom, nnode);
    gat_gemm  <<<(nnode + 15) / 16, 256, 0, stream>>>(h, W, Wh, nnode);
    gat_proj  <<<(nnode + 255) / 256, 256, 0, stream>>>(Wh, a, Wh1, Wh2, nnode);
    gat_logits<<<(nedge + 255) / 256, 256, 0, stream>>>(src, dst, Wh1, Wh2, ebuf, mmax, nedge);
    gat_exp   <<<(nedge + 255) / 256, 256, 0, stream>>>(src, ebuf, mmax, denom, nedge);
    {
        long long threads = (long long)nedge * 32;
        gat_agg<<<(unsigned)((threads + 255) / 256), 256, 0, stream>>>(src, dst, ebuf, Wh, out, nedge);
    }
    gat_final <<<(nOut + 255) / 256, 256, 0, stream>>>(out, denom, nOut);
}